// memory_57475252355409
// MI455X (gfx1250) — compile-verified
//
#include <hip/hip_runtime.h>
#include <hip/hip_bf16.h>

typedef __attribute__((ext_vector_type(2))) float v2f;
typedef __attribute__((ext_vector_type(8))) float v8f;

#define HEADS 4
#define KD2 256
#define BLOCKN 128
#define COLN 512
#define D_OUT 512
#define D_IN 1024
#define QUERYDIM 1024
#define T_LEN 2048
#define B_SZ 2
#define TOPK 256   // T*KNN/BLOCK = 2048*16/128

// ---------------------------------------------------------------- utilities
__global__ void zero_f4_kernel(float4* __restrict__ p, int n4) {
    int i = blockIdx.x * blockDim.x + threadIdx.x;
    if (i < n4) p[i] = make_float4(0.f, 0.f, 0.f, 0.f);
}

// ------------------------------------------- causal depthwise conv (K=3) + bias
// y[b,t,c] = w[c,0]*x[b,t-2,c] + w[c,1]*x[b,t-1,c] + w[c,2]*x[b,t,c] + b[c]
__global__ void conv_kernel(const float* __restrict__ x, const float* __restrict__ w,
                            const float* __restrict__ bias, float* __restrict__ y, int total) {
    int i = blockIdx.x * blockDim.x + threadIdx.x;
    if (i >= total) return;
    int c = i & (D_IN - 1);
    int t = (i >> 10) & (T_LEN - 1);
    float acc = bias[c] + x[i] * w[c * 3 + 2];
    if (t >= 1) acc += x[i - D_IN] * w[c * 3 + 1];
    if (t >= 2) acc += x[i - 2 * D_IN] * w[c * 3 + 0];
    y[i] = acc;
}

// ------------------------------------------------- fp32 WMMA GEMM, C = A * B^T
// Register-blocked: one 32x32 output block per wave = 2x2 grid of 16x16 WMMA
// tiles, 4 v8f accumulators. Per K-step of 4: 2 A-frags + 2 B-frags feed 4
// v_wmma_f32_16x16x4_f32 -> 8 FLOP/byte of VMEM (operands are L2-resident).
// A is (M,K) row-major, ldA; B-operand accessed as Bt[n*ldB + k] which matches
// lin_w[n,k], rowkeys[e,h,k], colkeys[c,h,k] with no repacking.
// gridDim.z batches (b*H+h): A += z*Az, Bt += (z%bMod)*Bz, C += z*Cz.
// V_WMMA_F32_16X16X4_F32 lane layout (ISA 7.12.2):
//   A 16x4 : lane&15 = M ; v0 = K(2*(lane>>4)), v1 = K(2*(lane>>4)+1)
//   B 4x16 : lane&15 = N ; same K striping
//   C 16x16: v[j] -> row = j + 8*(lane>>4), col = lane&15
__global__ void gemm_wmma_f32(const float* __restrict__ A, int ldA, long Az,
                              const float* __restrict__ Bt, int ldB, long Bz, int bMod,
                              float* __restrict__ C, int ldC, long Cz,
                              int M, int N, int K, const float* __restrict__ bias) {
    const int wave = threadIdx.x >> 5;
    const int lane = threadIdx.x & 31;
    const int half = lane >> 4;
    const int l15  = lane & 15;
    const int blkN = N >> 5;                                    // 32-wide blocks
    const int blk  = blockIdx.x * (blockDim.x >> 5) + wave;     // wave-uniform
    if (blk >= (M >> 5) * blkN) return;                         // EXEC all-1s inside
    const int bm = blk / blkN;
    const int bn = blk - bm * blkN;
    const int z = blockIdx.z;

    const float* A0 = A + (long)z * Az + (long)(bm * 32 + l15) * ldA;        // rows 0..15
    const float* A1 = A0 + (long)16 * ldA;                                   // rows 16..31
    const float* B0 = Bt + (long)(z % bMod) * Bz + (long)(bn * 32 + l15) * ldB;
    const float* B1 = B0 + (long)16 * ldB;

    v8f acc00 = {}, acc01 = {}, acc10 = {}, acc11 = {};
    const int kh = 2 * half;
    for (int k0 = 0; k0 < K; k0 += 4) {
        v2f a0, a1, b0, b1;
        a0.x = A0[k0 + kh]; a0.y = A0[k0 + kh + 1];
        a1.x = A1[k0 + kh]; a1.y = A1[k0 + kh + 1];
        b0.x = B0[k0 + kh]; b0.y = B0[k0 + kh + 1];
        b1.x = B1[k0 + kh]; b1.y = B1[k0 + kh + 1];
        acc00 = __builtin_amdgcn_wmma_f32_16x16x4_f32(false, a0, false, b0, (short)0, acc00, false, false);
        acc01 = __builtin_amdgcn_wmma_f32_16x16x4_f32(false, a0, false, b1, (short)0, acc01, false, false);
        acc10 = __builtin_amdgcn_wmma_f32_16x16x4_f32(false, a1, false, b0, (short)0, acc10, false, false);
        acc11 = __builtin_amdgcn_wmma_f32_16x16x4_f32(false, a1, false, b1, (short)0, acc11, false, false);
    }
    const float bv0 = bias ? bias[bn * 32 + l15] : 0.0f;
    const float bv1 = bias ? bias[bn * 32 + 16 + l15] : 0.0f;
    float* Crow = C + (long)z * Cz + (long)bn * 32 + l15;
#pragma unroll
    for (int j = 0; j < 8; ++j) {
        long r0 = (long)(bm * 32 + j + 8 * half) * ldC;
        long r1 = (long)(bm * 32 + 16 + j + 8 * half) * ldC;
        Crow[r0]      = acc00[j] + bv0;
        Crow[r0 + 16] = acc01[j] + bv1;
        Crow[r1]      = acc10[j] + bv0;
        Crow[r1 + 16] = acc11[j] + bv1;
    }
}

// ------------------------------------------- top-1 over COL per (b,h,t) row
__global__ void colmax_kernel(const float* __restrict__ colsc, float* __restrict__ score_col,
                              int* __restrict__ idx_col, int rows) {
    int w = (blockIdx.x * blockDim.x + threadIdx.x) >> 5;
    int lane = threadIdx.x & 31;
    if (w >= rows) return;
    const float* row = colsc + (long)w * COLN;
    float best = -INFINITY;
    int bi = COLN;
    for (int i = lane; i < COLN; i += 32) {
        float v = row[i];
        if (v > best) { best = v; bi = i; }   // first strictly-greater keeps lowest index
    }
    for (int off = 16; off > 0; off >>= 1) {  // wave32 reduction
        float ov = __shfl_xor(best, off, 32);
        int oi = __shfl_xor(bi, off, 32);
        if (ov > best || (ov == best && oi < bi)) { best = ov; bi = oi; }
    }
    if (lane == 0) { score_col[w] = best; idx_col[w] = bi; }
}

// ---- exact top-256-of-2048 per (b,h,r): bitonic sort in LDS, desc, index tiebreak
// score2[b,h,r,tt] = score[b,h, r*16 + tt>>7, tt&127] + score_col[b,h, r*16 + tt>>7]
__global__ void topk_kernel(const float* __restrict__ score, const float* __restrict__ score_col,
                            const int* __restrict__ idx_col,
                            float* __restrict__ score_top, int* __restrict__ disp,
                            int* __restrict__ idxt) {
    __shared__ float sv[T_LEN];
    __shared__ int   si[T_LEN];
    const int bhr = blockIdx.x;
    const int r   = bhr & (BLOCKN - 1);
    const int bh  = bhr >> 7;
    const int tid = threadIdx.x;

    for (int tt = tid; tt < T_LEN; tt += blockDim.x) {
        int t = (r << 4) + (tt >> 7);
        int e = tt & (BLOCKN - 1);
        sv[tt] = score[((long)bh * T_LEN + t) * BLOCKN + e] + score_col[bh * T_LEN + t];
        si[tt] = tt;
    }
    for (int size = 2; size <= T_LEN; size <<= 1) {
        for (int stride = size >> 1; stride > 0; stride >>= 1) {
            __syncthreads();
            for (int p = tid; p < T_LEN / 2; p += blockDim.x) {
                int lo = 2 * p - (p & (stride - 1));
                int hi = lo + stride;
                float vl = sv[lo], vh = sv[hi];
                int il = si[lo], ih = si[hi];
                // "hi element goes first in descending order" (value desc, index asc on ties)
                bool hiBefore = (vh > vl) || (vh == vl && ih < il);
                bool desc = ((lo & size) == 0);
                if (desc == hiBefore) { sv[lo] = vh; sv[hi] = vl; si[lo] = ih; si[hi] = il; }
            }
        }
    }
    __syncthreads();
    if (tid < TOPK) {
        float v = sv[tid];
        int tt = si[tid];
        int t = (r << 4) + (tt >> 7);
        int e = tt & (BLOCKN - 1);
        long o = (long)bhr * TOPK + tid;
        score_top[o] = v;
        disp[o] = tt;                                   // token index in [0,T)
        idxt[o] = e * COLN + idx_col[bh * T_LEN + t];   // row of value_w
    }
}

// ---------------- gather value_w row, scale, scatter-add into out (atomics)
__global__ void scatter_kernel(const float* __restrict__ score_top, const int* __restrict__ disp,
                               const int* __restrict__ idxt, const float* __restrict__ value_w,
                               float* __restrict__ out, int nEntries) {
    int entry = (blockIdx.x * blockDim.x + threadIdx.x) >> 5;  // one wave per entry
    int lane = threadIdx.x & 31;
    if (entry >= nEntries) return;
    float s = score_top[entry];
    int tt = disp[entry];
    int row = idxt[entry];
    int b = entry / (HEADS * BLOCKN * TOPK);
    const float4* vrow = (const float4*)(value_w + (long)row * D_OUT);
    float* orow = out + ((long)b * T_LEN + tt) * D_OUT;
    __builtin_prefetch(vrow, 0, 1);   // global_prefetch_b8; value_w (134MB) fits in 192MB L2
#pragma unroll
    for (int it = 0; it < 4; ++it) {
        int i = it * 32 + lane;       // float4 index within 128 float4s of the row
        float4 v = vrow[i];
        int o = i * 4;
        atomicAdd(&orow[o + 0], s * v.x);
        atomicAdd(&orow[o + 1], s * v.y);
        atomicAdd(&orow[o + 2], s * v.z);
        atomicAdd(&orow[o + 3], s * v.w);
    }
}

extern "C" void kernel_launch(void* const* d_in, const int* in_sizes, int n_in,
                              void* d_out, int out_size, void* d_ws, size_t ws_size,
                              hipStream_t stream) {
    const float* x       = (const float*)d_in[0];
    const float* conv_w  = (const float*)d_in[1];
    const float* conv_b  = (const float*)d_in[2];
    const float* lin_w   = (const float*)d_in[3];
    const float* lin_b   = (const float*)d_in[4];
    const float* rowkeys = (const float*)d_in[5];
    const float* colkeys = (const float*)d_in[6];
    const float* value_w = (const float*)d_in[7];
    float* out = (float*)d_out;

    // ---- carve workspace (~79 MB total)
    char* ws = (char*)d_ws;
    size_t off = 0;
    auto carve = [&](size_t bytes) -> void* {
        void* p = ws + off;
        off += (bytes + 255) & ~(size_t)255;
        return p;
    };
    float* y         = (float*)carve((size_t)B_SZ * T_LEN * D_IN * 4);
    float* query     = (float*)carve((size_t)B_SZ * T_LEN * QUERYDIM * 4);
    float* score     = (float*)carve((size_t)B_SZ * HEADS * T_LEN * BLOCKN * 4);
    float* colsc     = (float*)carve((size_t)B_SZ * HEADS * T_LEN * COLN * 4);
    float* score_col = (float*)carve((size_t)B_SZ * HEADS * T_LEN * 4);
    int*   idx_col   = (int*)  carve((size_t)B_SZ * HEADS * T_LEN * 4);
    float* score_top = (float*)carve((size_t)B_SZ * HEADS * BLOCKN * TOPK * 4);
    int*   dispb     = (int*)  carve((size_t)B_SZ * HEADS * BLOCKN * TOPK * 4);
    int*   idxtb     = (int*)  carve((size_t)B_SZ * HEADS * BLOCKN * TOPK * 4);

    // 0) zero output (scatter-add target)
    {
        int n4 = B_SZ * T_LEN * D_OUT / 4;
        zero_f4_kernel<<<(n4 + 255) / 256, 256, 0, stream>>>((float4*)out, n4);
    }
    // 1) causal depthwise conv + bias
    {
        int total = B_SZ * T_LEN * D_IN;
        conv_kernel<<<total / 256, 256, 0, stream>>>(x, conv_w, conv_b, y, total);
    }
    // 2) query = y @ lin_w^T + lin_b   (M=4096, N=1024, K=1024), fp32 WMMA
    {
        int M = B_SZ * T_LEN, N = QUERYDIM, K = D_IN;
        int blks = (M / 32) * (N / 32);
        dim3 grid((blks + 7) / 8, 1, 1);
        gemm_wmma_f32<<<grid, 256, 0, stream>>>(y, K, 0, lin_w, D_IN, 0, 1,
                                                query, N, 0, M, N, K, lin_b);
    }
    // Raw reshape (B,T,1024)->(B,H,T,256): Q_bh = query + (b*H+h)*T*KD2, row stride KD2.
    // 3) score[b,h,t,e] = Q_bh @ rowkeys[:,h,:]^T   (z = b*H+h, 8 batches)
    {
        int M = T_LEN, N = BLOCKN, K = KD2;
        int blks = (M / 32) * (N / 32);
        dim3 grid((blks + 7) / 8, 1, B_SZ * HEADS);
        gemm_wmma_f32<<<grid, 256, 0, stream>>>(query, KD2, (long)T_LEN * KD2,
                                                rowkeys, HEADS * KD2, KD2, HEADS,
                                                score, BLOCKN, (long)T_LEN * BLOCKN,
                                                M, N, K, nullptr);
    }
    // 4) col_scores[b,h,t,c] = Q_bh @ colkeys[:,h,:]^T
    {
        int M = T_LEN, N = COLN, K = KD2;
        int blks = (M / 32) * (N / 32);
        dim3 grid((blks + 7) / 8, 1, B_SZ * HEADS);
        gemm_wmma_f32<<<grid, 256, 0, stream>>>(query, KD2, (long)T_LEN * KD2,
                                                colkeys, HEADS * KD2, KD2, HEADS,
                                                colsc, COLN, (long)T_LEN * COLN,
                                                M, N, K, nullptr);
    }
    // 5) top-1 over columns
    {
        int rows = B_SZ * HEADS * T_LEN;                 // 16384 waves
        colmax_kernel<<<(rows * 32 + 255) / 256, 256, 0, stream>>>(colsc, score_col, idx_col, rows);
    }
    // 6) exact top-256 of 2048 per (b,h,r)
    topk_kernel<<<B_SZ * HEADS * BLOCKN, 256, 0, stream>>>(score, score_col, idx_col,
                                                           score_top, dispb, idxtb);
    // 7) gather/scale/scatter-add
    {
        int nE = B_SZ * HEADS * BLOCKN * TOPK;           // 262144 entries
        scatter_kernel<<<(nE * 32 + 255) / 256, 256, 0, stream>>>(score_top, dispb, idxtb,
                                                                  value_w, out, nE);
    }
}